// MinibatchDiscrimination_1176821039198
// MI455X (gfx1250) — compile-verified
//
#include <hip/hip_runtime.h>

typedef __attribute__((ext_vector_type(2))) float v2f;
typedef __attribute__((ext_vector_type(4))) float v4f;
typedef __attribute__((ext_vector_type(8))) float v8f;

#define BSZ   512
#define INF_  512          // IN
#define OUTF  64           // OUT
#define KD    16           // kernel dim
#define NCOL  (OUTF * KD)  // 1024 columns of T-as-matrix
#define OUTW  (INF_ + OUTF) // 576 output row width

// ---------------------------------------------------------------------------
// Kernel 1: copy x into the first 512 columns of each 576-wide output row.
// Vectorized float4: 512*512 floats = 65536 float4. 576 % 4 == 0 so the
// float4 view of the output rows stays aligned.
// ---------------------------------------------------------------------------
__global__ __launch_bounds__(256) void copy_x_kernel(const float* __restrict__ x,
                                                     float* __restrict__ out) {
    int idx = blockIdx.x * 256 + threadIdx.x;     // 0..65535
    int row = idx >> 7;                            // /128  (128 float4 per row)
    int c4  = idx & 127;
    const v4f* xv = (const v4f*)x;
    v4f* ov = (v4f*)out;
    ov[(size_t)row * (OUTW / 4) + c4] = xv[idx];
}

// ---------------------------------------------------------------------------
// Kernel 2: M = x @ T via V_WMMA_F32_16X16X4_F32 (full fp32 GEMM).
// Grid: 256 blocks x dim3(32,8): 8 waves/block, one 16x16 tile per wave,
// 2048 tiles total (32 M-tiles x 64 N-tiles). One N-tile == one 'o' slab,
// so D is stored directly as Mws[o][i][k] (o-major) for the pairwise stage.
//
// A fragment (16x4 f32, 2 VGPRs): lane l holds row M=l&15,
//   K = kk + (l>>4)*2 + {0,1}  -> one b64 load.
// B fragment (4x16 f32, 2 VGPRs): lane l holds col N=l&15,
//   K = kk + (l>>4)*2 + {0,1}  -> two b32 loads (stride = 1024 floats).
// ---------------------------------------------------------------------------
__global__ __launch_bounds__(256) void gemm_wmma_kernel(const float* __restrict__ x,
                                                        const float* __restrict__ T,
                                                        float* __restrict__ Mws) {
    const int lane = threadIdx.x;                 // 0..31 (wave32)
    const int wave = threadIdx.y;                 // 0..7
    const int tile = blockIdx.x * 8 + wave;       // 0..2047
    const int mT = tile >> 6;                     // 0..31  (row tile)
    const int nT = tile & 63;                     // 0..63  (== o)

    const int half = lane >> 4;                   // 0 or 1
    const int lm   = lane & 15;

    const float* aPtr = x + (size_t)(mT * 16 + lm) * INF_;  // A row base
    const float* bPtr = T + (size_t)(nT * 16 + lm);         // B column base

    v8f c = {};
    for (int kk = 0; kk < INF_; kk += 4) {
        const int ka = kk + half * 2;             // even -> 8B aligned
        v2f a = *(const v2f*)(aPtr + ka);         // global_load_b64
        v2f b;
        b.x = bPtr[(size_t)ka * NCOL];
        b.y = bPtr[(size_t)(ka + 1) * NCOL];
        // 8 args: (neg_a, A, neg_b, B, c_mod, C, reuse_a, reuse_b)
        c = __builtin_amdgcn_wmma_f32_16x16x4_f32(
                /*neg_a=*/false, a, /*neg_b=*/false, b,
                /*c_mod=*/(short)0, c, /*reuse_a=*/false, /*reuse_b=*/false);
    }

    // D layout: vgpr v, lanes 0-15 -> (i_local=v, k=lm); lanes 16-31 -> (v+8, lm)
    float* dPtr = Mws + ((size_t)nT * BSZ + mT * 16) * KD;
#pragma unroll
    for (int v = 0; v < 8; ++v) {
        const int iLocal = v + half * 8;
        dPtr[iLocal * KD + lm] = c[v];
    }
}

// ---------------------------------------------------------------------------
// Kernel 3: pairwise L1 + exp reduction. One block per o (64 blocks).
// Stage the 512x16 slab (32 KB) in LDS; inner loop reads S[j][*] with every
// lane at the same address -> LDS broadcast, conflict-free. Each thread keeps
// 2 rows (i, i+256) in registers so each broadcast is reused twice.
// ---------------------------------------------------------------------------
__global__ __launch_bounds__(256) void pairwise_kernel(const float* __restrict__ Mws,
                                                       float* __restrict__ out) {
    __shared__ float S[BSZ * KD];                 // 32 KB
    const int o   = blockIdx.x;                   // 0..63
    const int tid = threadIdx.x;                  // 0..255

    // Cooperative slab load: 8192 floats = 2048 float4 by 256 threads.
    const v4f* src = (const v4f*)(Mws + (size_t)o * BSZ * KD);
    v4f* dst = (v4f*)S;
#pragma unroll
    for (int t = 0; t < 8; ++t) dst[tid + 256 * t] = src[tid + 256 * t];
    __syncthreads();

    const int i0 = tid;
    const int i1 = tid + 256;
    float a0[KD], a1[KD];
#pragma unroll
    for (int k = 0; k < KD; ++k) {
        a0[k] = S[i0 * KD + k];
        a1[k] = S[i1 * KD + k];
    }

    float acc0 = 0.0f, acc1 = 0.0f;
    for (int j = 0; j < BSZ; ++j) {
        float n0 = 0.0f, n1 = 0.0f;
#pragma unroll
        for (int k = 0; k < KD; ++k) {
            const float s = S[j * KD + k];        // broadcast read
            n0 += fabsf(a0[k] - s);               // abs folds into add modifier
            n1 += fabsf(a1[k] - s);
        }
        acc0 += __expf(-n0);                      // v_exp_f32
        acc1 += __expf(-n1);
    }
    // self term exp(0)=1 is included in acc -> subtract 1 like the reference
    out[(size_t)i0 * OUTW + INF_ + o] = acc0 - 1.0f;
    out[(size_t)i1 * OUTW + INF_ + o] = acc1 - 1.0f;
}

// ---------------------------------------------------------------------------
extern "C" void kernel_launch(void* const* d_in, const int* in_sizes, int n_in,
                              void* d_out, int out_size, void* d_ws, size_t ws_size,
                              hipStream_t stream) {
    const float* x = (const float*)d_in[0];       // (512, 512) f32
    const float* T = (const float*)d_in[1];       // (512, 64, 16) f32
    float* out = (float*)d_out;                   // (512, 576) f32
    float* Mws = (float*)d_ws;                    // scratch: 512*64*16 f32 = 2 MB

    copy_x_kernel<<<256, 256, 0, stream>>>(x, out);
    gemm_wmma_kernel<<<256, dim3(32, 8), 0, stream>>>(x, T, Mws);
    pairwise_kernel<<<64, 256, 0, stream>>>(Mws, out);
}